// MemristorArray_67585605370369
// MI455X (gfx1250) — compile-verified
//
#include <hip/hip_runtime.h>
#include <hip/hip_bf16.h>

// ---------------------------------------------------------------------------
// MemristorArray forward on MI455X (gfx1250, wave32).
//   out[b,o] = sum_i L[b,i]                      (row sum)
//            + sum_i (H-L)[b,i] * r[i,o]         (f32 WMMA GEMM)
//            + sum_i noise[i,o]*sqrt(|rr|)*sg    (VALU/TRANS sweep)
// ---------------------------------------------------------------------------

typedef __attribute__((ext_vector_type(2))) float v2f;
typedef __attribute__((ext_vector_type(8))) float v8f;

#define B_DIM 128
#define I_DIM 512
#define O_DIM 512

// 4*k_B*T*BW and 2*e*BW
#define C1_JOHNSON 1.65677880e-28f
#define C2_SHOT    5.43656366e-08f
#define NOISE_MIN  1e-12f

// ---------------------------------------------------------------------------
// Kernel A: per-(b,i) precompute (Horner polys, sigma prefactor) + row sum.
// One block per batch row b; 256 threads each handle 2 of the 512 i's.
// ---------------------------------------------------------------------------
__global__ __launch_bounds__(256) void precomp_kernel(
    const float* __restrict__ x,      // [B,I]
    const float* __restrict__ pl,     // [7] ascending
    const float* __restrict__ ph,     // [6] ascending
    float* __restrict__ L,            // [B,I]
    float* __restrict__ Dm,           // [B,I]  (H - L)
    float* __restrict__ SG,           // [B,I]  sqrt(c1/(|x|+eps)+c2)
    float* __restrict__ rowSumL) {    // [B]
  __shared__ float red[256];
  const int b = blockIdx.x;
  // hoist coefficients into registers (uniform -> scalar cache)
  float c0 = pl[0], c1 = pl[1], c2 = pl[2], c3 = pl[3], c4 = pl[4], c5 = pl[5], c6 = pl[6];
  float d0 = ph[0], d1 = ph[1], d2 = ph[2], d3 = ph[3], d4 = ph[4], d5 = ph[5];
  float lsum = 0.0f;
  for (int i = threadIdx.x; i < I_DIM; i += 256) {
    const float xv = x[b * I_DIM + i];
    float l = c6;
    l = fmaf(l, xv, c5); l = fmaf(l, xv, c4); l = fmaf(l, xv, c3);
    l = fmaf(l, xv, c2); l = fmaf(l, xv, c1); l = fmaf(l, xv, c0);
    float h = d5;
    h = fmaf(h, xv, d4); h = fmaf(h, xv, d3); h = fmaf(h, xv, d2);
    h = fmaf(h, xv, d1); h = fmaf(h, xv, d0);
    const float g = C1_JOHNSON / (fabsf(xv) + NOISE_MIN) + C2_SHOT;
    L[b * I_DIM + i]  = l;
    Dm[b * I_DIM + i] = h - l;
    SG[b * I_DIM + i] = sqrtf(g);
    lsum += l;
  }
  red[threadIdx.x] = lsum;
  __syncthreads();
  for (int s = 128; s > 0; s >>= 1) {
    if (threadIdx.x < (unsigned)s) red[threadIdx.x] += red[threadIdx.x + s];
    __syncthreads();
  }
  if (threadIdx.x == 0) rowSumL[b] = red[0];
}

// ---------------------------------------------------------------------------
// Kernel B: reproduce jax.random.normal(jax.random.key(42), (I,O), f32).
// threefry2x32 with key (0,42); counters split [0..n/2) / [n/2..n);
// bits -> uniform in [nextafter(-1,0), 1) -> sqrt(2)*erfinv (Giles poly).
// ---------------------------------------------------------------------------
__device__ __forceinline__ unsigned rotl32(unsigned v, int r) {
  return (v << r) | (v >> (32 - r));
}

__device__ __forceinline__ float bits_to_normal(unsigned bits) {
  // uniform in [lo, hi) exactly as jax.random.uniform
  const float lo = -0.99999994f;  // nextafter(-1, 0) in f32
  const float hi = 1.0f;
  float f = __uint_as_float((bits >> 9) | 0x3f800000u) - 1.0f;  // [0,1)
  float u = fmaf(f, hi - lo, lo);
  u = fmaxf(u, lo);
  // Giles (2010) single-precision erfinv
  float w = -__logf((1.0f - u) * (1.0f + u));
  float p;
  if (w < 5.0f) {
    w -= 2.5f;
    p = 2.81022636e-08f;
    p = fmaf(p, w, 3.43273939e-07f);
    p = fmaf(p, w, -3.5233877e-06f);
    p = fmaf(p, w, -4.39150654e-06f);
    p = fmaf(p, w, 0.00021858087f);
    p = fmaf(p, w, -0.00125372503f);
    p = fmaf(p, w, -0.00417768164f);
    p = fmaf(p, w, 0.246640727f);
    p = fmaf(p, w, 1.50140941f);
  } else {
    w = sqrtf(w) - 3.0f;
    p = -0.000200214257f;
    p = fmaf(p, w, 0.000100950558f);
    p = fmaf(p, w, 0.00134934322f);
    p = fmaf(p, w, -0.00367342844f);
    p = fmaf(p, w, 0.00573950773f);
    p = fmaf(p, w, -0.0076224613f);
    p = fmaf(p, w, 0.00943887047f);
    p = fmaf(p, w, 1.00167406f);
    p = fmaf(p, w, 2.83297682f);
  }
  return 1.41421356237f * (p * u);
}

__global__ __launch_bounds__(256) void noise_kernel(float* __restrict__ N) {
  const unsigned half = (I_DIM * O_DIM) / 2;  // 131072
  const unsigned j = blockIdx.x * 256u + threadIdx.x;
  if (j >= half) return;
  unsigned x0 = j, x1 = j + half;
  const unsigned ks0 = 0u, ks1 = 42u;
  const unsigned ks2 = 0x1BD11BDAu ^ ks0 ^ ks1;
  x0 += ks0; x1 += ks1;
  const int ra[4] = {13, 15, 26, 6};
  const int rb[4] = {17, 29, 16, 24};
#pragma unroll
  for (int r = 0; r < 4; ++r) { x0 += x1; x1 = rotl32(x1, ra[r]); x1 ^= x0; }
  x0 += ks1; x1 += ks2 + 1u;
#pragma unroll
  for (int r = 0; r < 4; ++r) { x0 += x1; x1 = rotl32(x1, rb[r]); x1 ^= x0; }
  x0 += ks2; x1 += ks0 + 2u;
#pragma unroll
  for (int r = 0; r < 4; ++r) { x0 += x1; x1 = rotl32(x1, ra[r]); x1 ^= x0; }
  x0 += ks0; x1 += ks1 + 3u;
#pragma unroll
  for (int r = 0; r < 4; ++r) { x0 += x1; x1 = rotl32(x1, rb[r]); x1 ^= x0; }
  x0 += ks1; x1 += ks2 + 4u;
#pragma unroll
  for (int r = 0; r < 4; ++r) { x0 += x1; x1 = rotl32(x1, ra[r]); x1 ^= x0; }
  x0 += ks2; x1 += ks0 + 5u;
  N[j]        = bits_to_normal(x0);
  N[j + half] = bits_to_normal(x1);
}

// ---------------------------------------------------------------------------
// Kernel C: out[b,o] = rowSumL[b] + (D @ r)[b,o] via v_wmma_f32_16x16x4_f32.
// One wave per 16x16 C tile; 8 waves/block; K loop of 128 WMMA steps.
// f32 A 16x4 layout: lane l (m = l&15), half h = l>>4: vgpr j -> K = 2h + j.
// f32 B 4x16 layout: lane l (n = l&15): vgpr j -> K = 2h + j.
// f32 C/D: vgpr v -> (M = v + 8h, N = l&15).
// ---------------------------------------------------------------------------
__global__ __launch_bounds__(256) void gemm_wmma_kernel(
    const float* __restrict__ Dm,       // [B,I]
    const float* __restrict__ R,        // [I,O]
    const float* __restrict__ rowSumL,  // [B]
    float* __restrict__ out) {          // [B,O]
  const int lane  = threadIdx.x & 31;
  const int wave  = threadIdx.x >> 5;
  const int ntile = blockIdx.x * 8 + wave;   // O/16 tiles, grid.x = 4
  const int mtile = blockIdx.y;              // B/16 tiles, grid.y = 8
  const int m = mtile * 16 + (lane & 15);
  const int n = ntile * 16 + (lane & 15);
  const int h = lane >> 4;

  v8f c = {};
  for (int k0 = 0; k0 < I_DIM; k0 += 4) {
    const int ka = k0 + 2 * h;
    // A pair: D[m, ka], D[m, ka+1] — one 64-bit load (8B aligned)
    v2f a = *(const v2f*)(Dm + m * I_DIM + ka);
    v2f bm;
    bm.x = R[ka * O_DIM + n];
    bm.y = R[(ka + 1) * O_DIM + n];
    c = __builtin_amdgcn_wmma_f32_16x16x4_f32(
        /*neg_a=*/false, a, /*neg_b=*/false, bm,
        /*c_mod=*/(short)0, c, /*reuse_a=*/false, /*reuse_b=*/false);
  }
#pragma unroll
  for (int v = 0; v < 8; ++v) {
    const int row = mtile * 16 + v + 8 * h;
    out[row * O_DIM + n] = c[v] + rowSumL[row];
  }
}

// ---------------------------------------------------------------------------
// Kernel D: out[b,o] += sum_i noise[i,o] * sqrt(|L + D*r|) * sg[b,i].
// 256 threads own an 8(b) x 32(o) tile; L/D/sg staged in LDS (broadcast
// reads), r/noise coalesced from L1/L2 (1 MB each -> resident).
// ---------------------------------------------------------------------------
#define TB_B 8
#define TB_O 32
#define ICHUNK 64

__global__ __launch_bounds__(256) void noise_accum_kernel(
    const float* __restrict__ L, const float* __restrict__ Dm,
    const float* __restrict__ SG, const float* __restrict__ R,
    const float* __restrict__ N, float* __restrict__ out) {
  __shared__ float sL[TB_B][ICHUNK];
  __shared__ float sD[TB_B][ICHUNK];
  __shared__ float sG[TB_B][ICHUNK];
  const int t  = threadIdx.x;
  const int ob = t & (TB_O - 1);
  const int bb = t >> 5;                       // wave-uniform row
  const int o  = blockIdx.x * TB_O + ob;
  const int b  = blockIdx.y * TB_B + bb;

  float acc = 0.0f;
  for (int i0 = 0; i0 < I_DIM; i0 += ICHUNK) {
    __syncthreads();
    for (int s = t; s < TB_B * ICHUNK; s += 256) {
      const int rb = s / ICHUNK, ii = s % ICHUNK;
      const int g  = (blockIdx.y * TB_B + rb) * I_DIM + i0 + ii;
      sL[rb][ii] = L[g];
      sD[rb][ii] = Dm[g];
      sG[rb][ii] = SG[g];
    }
    __syncthreads();
    if (i0 + ICHUNK < I_DIM) {  // prime next r/noise chunk
      __builtin_prefetch(&R[(i0 + ICHUNK) * O_DIM + o], 0, 0);
      __builtin_prefetch(&N[(i0 + ICHUNK) * O_DIM + o], 0, 0);
    }
#pragma unroll 4
    for (int ii = 0; ii < ICHUNK; ++ii) {
      const int i  = i0 + ii;
      const float rv = R[i * O_DIM + o];
      const float nv = N[i * O_DIM + o];
      const float rr = fmaf(sD[bb][ii], rv, sL[bb][ii]);
      acc = fmaf(nv * sG[bb][ii], sqrtf(fabsf(rr)), acc);
    }
  }
  out[b * O_DIM + o] += acc;  // kernel C already wrote the base value
}

// ---------------------------------------------------------------------------
extern "C" void kernel_launch(void* const* d_in, const int* in_sizes, int n_in,
                              void* d_out, int out_size, void* d_ws, size_t ws_size,
                              hipStream_t stream) {
  (void)in_sizes; (void)n_in; (void)out_size; (void)ws_size;
  const float* x  = (const float*)d_in[0];   // [128,512]
  const float* pl = (const float*)d_in[1];   // [7]
  const float* ph = (const float*)d_in[2];   // [6]
  const float* r  = (const float*)d_in[3];   // [512,512]
  float* out = (float*)d_out;                // [128,512]

  float* wL  = (float*)d_ws;                 // 65536
  float* wD  = wL + B_DIM * I_DIM;           // 65536
  float* wS  = wD + B_DIM * I_DIM;           // 65536
  float* wN  = wS + B_DIM * I_DIM;           // 262144
  float* wRS = wN + I_DIM * O_DIM;           // 128

  // A: per-(b,i) precompute + row sums of L
  precomp_kernel<<<dim3(B_DIM), dim3(256), 0, stream>>>(x, pl, ph, wL, wD, wS, wRS);

  // B: exact-threefry noise field [I,O]
  noise_kernel<<<dim3((I_DIM * O_DIM / 2 + 255) / 256), dim3(256), 0, stream>>>(wN);

  // C: WMMA f32 GEMM (D @ r) + rowSumL -> out
  gemm_wmma_kernel<<<dim3(O_DIM / (16 * 8), B_DIM / 16), dim3(256), 0, stream>>>(
      wD, r, wRS, out);

  // D: accumulate noise * sigma into out
  noise_accum_kernel<<<dim3(O_DIM / TB_O, B_DIM / TB_B), dim3(256), 0, stream>>>(
      wL, wD, wS, r, wN, out);
}